// Set2SetThenCat_18640158064910
// MI455X (gfx1250) — compile-verified
//
#include <hip/hip_runtime.h>
#include <hip/hip_bf16.h>

// ---------------- problem constants ----------------
#define DD     128          // feature dim
#define BB     1024         // number of segments / batch
#define NATOM  500000
#define NBOND2 1000000
#define NBONDV (NBOND2/2)   // bond view rows
#define GATES  (4*DD)       // 512
#define KTOT   (3*DD)       // 384  = [q_star(256) | h(128)]

typedef __attribute__((ext_vector_type(2))) float v2f;
typedef __attribute__((ext_vector_type(8))) float v8f;

// ---------------- helpers ----------------
__global__ void zero_f32(float* p, int n) {
    int t = blockIdx.x * blockDim.x + threadIdx.x;
    if (t < n) p[t] = 0.0f;
}

// exclusive prefix sum of sizes/div -> offs[0..nseg], serial (nseg=1024, trivial)
__global__ void build_offsets(const int* __restrict__ sizes, int* __restrict__ offs,
                              int nseg, int div) {
    if (blockIdx.x == 0 && threadIdx.x == 0) {
        int acc = 0;
        for (int i = 0; i < nseg; ++i) { offs[i] = acc; acc += sizes[i] / div; }
        offs[nseg] = acc;
    }
}

// ---------------- LSTM gate GEMM via WMMA f32 16x16x4 ----------------
// gates(1024x512) = X(1024x384) * W^T(384x512), X = [q_star | h], W = [W_ih | W_hh]
__device__ __forceinline__ float ldX(const float* qs, const float* h, int m, int k) {
    return (k < 2*DD) ? qs[(size_t)m * (2*DD) + k]
                      : h [(size_t)m * DD + (k - 2*DD)];
}
__device__ __forceinline__ float ldW(const float* wih, const float* whh, int n, int k) {
    return (k < 2*DD) ? wih[(size_t)n * (2*DD) + k]
                      : whh[(size_t)n * DD + (k - 2*DD)];
}

__global__ void lstm_gemm_wmma(const float* __restrict__ qs, const float* __restrict__ h,
                               const float* __restrict__ wih, const float* __restrict__ whh,
                               float* __restrict__ gates) {
    const int lane   = threadIdx.x & 31;
    const int waveId = blockIdx.x * (blockDim.x >> 5) + (threadIdx.x >> 5);
    const int n_tile = waveId & 31;   // 512/16 = 32
    const int m_tile = waveId >> 5;   // 1024/16 = 64
    const int l15    = lane & 15;
    const int khalf  = (lane < 16) ? 0 : 2;   // ISA 16x4 f32 A/B lane split

    const int mrow = m_tile * 16 + l15;
    const int ncol = n_tile * 16 + l15;

    v8f acc = {};
    for (int k = 0; k < KTOT; k += 4) {
        v2f a, b;
        a[0] = ldX(qs, h, mrow, k + khalf + 0);
        a[1] = ldX(qs, h, mrow, k + khalf + 1);
        b[0] = ldW(wih, whh, ncol, k + khalf + 0);
        b[1] = ldW(wih, whh, ncol, k + khalf + 1);
        acc = __builtin_amdgcn_wmma_f32_16x16x4_f32(
            /*neg_a=*/false, a, /*neg_b=*/false, b,
            /*c_mod=*/(short)0, acc, /*reuse_a=*/false, /*reuse_b=*/false);
    }
    // D layout: VGPR r, lanes 0-15 -> (M=r, N=lane); lanes 16-31 -> (M=8+r, N=lane-16)
    const int mbase = m_tile * 16 + ((lane < 16) ? 0 : 8);
    #pragma unroll
    for (int r = 0; r < 8; ++r)
        gates[(size_t)(mbase + r) * GATES + ncol] = acc[r];
}

// ---------------- LSTM elementwise update ----------------
__global__ void lstm_elem(const float* __restrict__ gates,
                          const float* __restrict__ b_ih, const float* __restrict__ b_hh,
                          float* __restrict__ h, float* __restrict__ c,
                          float* __restrict__ qs) {
    int t = blockIdx.x * blockDim.x + threadIdx.x;
    if (t >= BB * DD) return;
    int b = t >> 7, j = t & (DD - 1);
    const float* g = gates + (size_t)b * GATES;
    float gi = g[j]          + b_ih[j]          + b_hh[j];
    float gf = g[DD + j]     + b_ih[DD + j]     + b_hh[DD + j];
    float gg = g[2*DD + j]   + b_ih[2*DD + j]   + b_hh[2*DD + j];
    float go = g[3*DD + j]   + b_ih[3*DD + j]   + b_hh[3*DD + j];
    float si = 1.0f / (1.0f + __expf(-gi));
    float sf = 1.0f / (1.0f + __expf(-gf));
    float so = 1.0f / (1.0f + __expf(-go));
    float cn = sf * c[t] + si * tanhf(gg);
    float hn = so * tanhf(cn);
    c[t] = cn;
    h[t] = hn;
    qs[(size_t)b * (2*DD) + j] = hn;   // q half of q_star
}

// ---------------- fused segment-softmax attention + readout ----------------
// one 128-thread block (4 waves) per segment; rowStride = 128 (atoms) / 256 (bond view)
__global__ void set2set_attn(const float* __restrict__ feat, int rowStride,
                             const int* __restrict__ offs,
                             const float* __restrict__ h /* = q */,
                             float* __restrict__ qs, float* __restrict__ e_ws) {
    const int b     = blockIdx.x;
    const int start = offs[b];
    const int end   = offs[b + 1];
    const int lane  = threadIdx.x & 31;
    const int wave  = threadIdx.x >> 5;

    __shared__ float s_red[4];
    __shared__ float s_acc[4][DD];

    const float4 q4 = *reinterpret_cast<const float4*>(h + (size_t)b * DD + 4 * lane);

    // pass A: e[row] = feat[row].q ; running (wave-uniform) max
    float mloc = -__builtin_inff();
    for (int r = start + wave; r < end; r += 4) {
        const float4 f4 = *reinterpret_cast<const float4*>(feat + (size_t)r * rowStride + 4 * lane);
        float p = f4.x * q4.x + f4.y * q4.y + f4.z * q4.z + f4.w * q4.w;
        #pragma unroll
        for (int off = 16; off > 0; off >>= 1) p += __shfl_down(p, off);
        p = __shfl(p, 0);
        if (lane == 0) e_ws[r] = p;
        mloc = fmaxf(mloc, p);
    }
    if (lane == 0) s_red[wave] = mloc;
    __syncthreads();
    const float m = fmaxf(fmaxf(s_red[0], s_red[1]), fmaxf(s_red[2], s_red[3]));
    __syncthreads();

    // pass B: weighted accumulation + denominator
    float4 acc = {0.f, 0.f, 0.f, 0.f};
    float denom = 0.f;
    for (int r = start + wave; r < end; r += 4) {
        const float ex = __expf(e_ws[r] - m);
        denom += ex;
        const float4 f4 = *reinterpret_cast<const float4*>(feat + (size_t)r * rowStride + 4 * lane);
        acc.x += f4.x * ex; acc.y += f4.y * ex;
        acc.z += f4.z * ex; acc.w += f4.w * ex;
    }
    s_acc[wave][4 * lane + 0] = acc.x;
    s_acc[wave][4 * lane + 1] = acc.y;
    s_acc[wave][4 * lane + 2] = acc.z;
    s_acc[wave][4 * lane + 3] = acc.w;
    if (lane == 0) s_red[wave] = denom;
    __syncthreads();

    const int t = threadIdx.x;   // column 0..127
    const float tot  = s_acc[0][t] + s_acc[1][t] + s_acc[2][t] + s_acc[3][t];
    const float dtot = s_red[0] + s_red[1] + s_red[2] + s_red[3];
    qs[(size_t)b * (2*DD) + DD + t] = (end > start) ? (tot / dtot) : 0.0f;
}

// ---------------- final concat: [qs_atom(256) | qs_bond(256) | feat_global(128)] ----------------
__global__ void concat_out(const float* __restrict__ qsa, const float* __restrict__ qsb,
                           const float* __restrict__ fg, float* __restrict__ out) {
    int t = blockIdx.x * blockDim.x + threadIdx.x;
    if (t >= BB * 640) return;
    int b = t / 640, j = t - b * 640;
    float v;
    if (j < 256)      v = qsa[(size_t)b * 256 + j];
    else if (j < 512) v = qsb[(size_t)b * 256 + (j - 256)];
    else              v = fg [(size_t)b * 128 + (j - 512)];
    out[t] = v;
}

// ---------------- host orchestration ----------------
extern "C" void kernel_launch(void* const* d_in, const int* in_sizes, int n_in,
                              void* d_out, int out_size, void* d_ws, size_t ws_size,
                              hipStream_t stream) {
    const float* feat_atom = (const float*)d_in[0];
    const float* feat_bond = (const float*)d_in[1];
    const float* feat_glob = (const float*)d_in[2];
    const float* aW_ih = (const float*)d_in[3];
    const float* aW_hh = (const float*)d_in[4];
    const float* ab_ih = (const float*)d_in[5];
    const float* ab_hh = (const float*)d_in[6];
    const float* bW_ih = (const float*)d_in[7];
    const float* bW_hh = (const float*)d_in[8];
    const float* bb_ih = (const float*)d_in[9];
    const float* bb_hh = (const float*)d_in[10];
    const int* sizes_atom = (const int*)d_in[11];
    const int* sizes_bond = (const int*)d_in[12];

    // -------- workspace layout --------
    char* ws = (char*)d_ws;
    size_t off = 0;
    auto carve = [&](size_t bytes) { void* p = ws + off; off = (off + bytes + 255) & ~(size_t)255; return p; };
    int*   off_a   = (int*)  carve((BB + 1) * sizeof(int));
    int*   off_b   = (int*)  carve((BB + 1) * sizeof(int));
    // per-set state: [h | c | q_star] contiguous so one zero kernel covers it
    float* state_a = (float*)carve((size_t)(BB*DD + BB*DD + BB*2*DD) * sizeof(float));
    float* state_b = (float*)carve((size_t)(BB*DD + BB*DD + BB*2*DD) * sizeof(float));
    float* gates_a = (float*)carve((size_t)BB * GATES * sizeof(float));
    float* gates_b = (float*)carve((size_t)BB * GATES * sizeof(float));
    float* e_a     = (float*)carve((size_t)NATOM  * sizeof(float));
    float* e_b     = (float*)carve((size_t)NBONDV * sizeof(float));
    (void)ws_size; (void)n_in; (void)in_sizes; (void)out_size;

    float* h_a = state_a;            float* c_a = h_a + BB*DD;  float* qs_a = c_a + BB*DD;
    float* h_b = state_b;            float* c_b = h_b + BB*DD;  float* qs_b = c_b + BB*DD;

    // -------- setup --------
    build_offsets<<<1, 1, 0, stream>>>(sizes_atom, off_a, BB, 1);
    build_offsets<<<1, 1, 0, stream>>>(sizes_bond, off_b, BB, 2);
    {
        const int nst = BB*DD + BB*DD + BB*2*DD;   // 524288
        zero_f32<<<(nst + 255) / 256, 256, 0, stream>>>(state_a, nst);
        zero_f32<<<(nst + 255) / 256, 256, 0, stream>>>(state_b, nst);
    }

    const int gemmBlocks = (64 * 32) / 8;          // 2048 waves / 8 waves-per-block
    const int elemBlocks = (BB * DD + 255) / 256;

    for (int it = 0; it < 3; ++it) {
        // ---- atoms ----
        lstm_gemm_wmma<<<gemmBlocks, 256, 0, stream>>>(qs_a, h_a, aW_ih, aW_hh, gates_a);
        lstm_elem<<<elemBlocks, 256, 0, stream>>>(gates_a, ab_ih, ab_hh, h_a, c_a, qs_a);
        set2set_attn<<<BB, 128, 0, stream>>>(feat_atom, DD, off_a, h_a, qs_a, e_a);
        // ---- bonds (even rows => rowStride 2*D) ----
        lstm_gemm_wmma<<<gemmBlocks, 256, 0, stream>>>(qs_b, h_b, bW_ih, bW_hh, gates_b);
        lstm_elem<<<elemBlocks, 256, 0, stream>>>(gates_b, bb_ih, bb_hh, h_b, c_b, qs_b);
        set2set_attn<<<BB, 128, 0, stream>>>(feat_bond, 2*DD, off_b, h_b, qs_b, e_b);
    }

    concat_out<<<(BB * 640 + 255) / 256, 256, 0, stream>>>(qs_a, qs_b, feat_glob, (float*)d_out);
}